// NodeBlock_74285754352302
// MI455X (gfx1250) — compile-verified
//
#include <hip/hip_runtime.h>

#define D_EDGE 64
#define D_NODE 128
#define D_OUT  128
#define D_IN   256   // 2*D_EDGE + D_NODE

typedef __attribute__((ext_vector_type(2))) float v2f;
typedef __attribute__((ext_vector_type(8))) float v8f;

// ---------------------------------------------------------------------------
// Kernel 1: zero the aggregation workspace (atomics accumulate into it).
// n is a multiple of 4 (2 * N * 64), use b128 stores.
// ---------------------------------------------------------------------------
__global__ void zero_f32_kernel(float4* __restrict__ p, long n4) {
  long i = (long)blockIdx.x * blockDim.x + threadIdx.x;
  long stride = (long)gridDim.x * blockDim.x;
  const float4 z = {0.0f, 0.0f, 0.0f, 0.0f};
  for (; i < n4; i += stride) p[i] = z;
}

// ---------------------------------------------------------------------------
// Kernel 2: scatter-sum edge features into node buffer (segment_sum).
// One thread per (edge, feature) element; coalesced NT loads (data is
// streamed exactly once -> don't rinse L2), coalesced f32 atomics.
// ---------------------------------------------------------------------------
__global__ void scatter_add_kernel(const float* __restrict__ edata,
                                   const long long* __restrict__ recv,
                                   float* __restrict__ agg, long n_edges) {
  long total = n_edges * (long)D_EDGE;
  long i = (long)blockIdx.x * blockDim.x + threadIdx.x;
  long stride = (long)gridDim.x * blockDim.x;
  for (; i < total; i += stride) {
    long e = i >> 6;            // i / D_EDGE
    int  d = (int)(i & 63);     // i % D_EDGE
    long r = (long)__builtin_nontemporal_load(&recv[e]);
    float v = __builtin_nontemporal_load(&edata[i]);
    atomicAdd(&agg[r * D_EDGE + d], v);
  }
}

// ---------------------------------------------------------------------------
// Kernel 3: fused concat + GEMM via V_WMMA_F32_16X16X4_F32.
// Block = 256 threads (8 wave32). Each block handles a 16-row stripe of x;
// wave w computes the 16x16 output tile at columns [16w, 16w+16).
// x = [agg_a | agg_b | vdata] assembled in LDS (row-padded vs bank conflicts).
// ---------------------------------------------------------------------------
__global__ __launch_bounds__(256) void node_gemm_wmma(
    const float* __restrict__ agg_a, const float* __restrict__ agg_b,
    const float* __restrict__ vdata, const float* __restrict__ W,
    const float* __restrict__ bias, float* __restrict__ out, int n_nodes) {
  __shared__ float xs[16][D_IN + 4];   // +4 floats pad: row stride 260 -> bank spread

  const int m0  = blockIdx.x * 16;
  const int tid = threadIdx.x;
  const bool full_tile = (m0 + 16) <= n_nodes;

  // Cooperative assembly of the 16x256 x-stripe into LDS.
  if (full_tile) {
#pragma unroll
    for (int it = 0; it < 16; ++it) {
      int idx  = tid + 256 * it;
      int row  = idx >> 8;     // / 256
      int col  = idx & 255;    // % 256
      int grow = m0 + row;
      float v;
      if (col < D_EDGE)            v = agg_a[(long)grow * D_EDGE + col];
      else if (col < 2 * D_EDGE)   v = agg_b[(long)grow * D_EDGE + (col - D_EDGE)];
      else                         v = vdata[(long)grow * D_NODE + (col - 2 * D_EDGE)];
      xs[row][col] = v;
    }
  } else {
    for (int idx = tid; idx < 16 * D_IN; idx += 256) {
      int row = idx >> 8;
      int col = idx & 255;
      int grow = m0 + row;
      float v = 0.0f;
      if (grow < n_nodes) {
        if (col < D_EDGE)            v = agg_a[(long)grow * D_EDGE + col];
        else if (col < 2 * D_EDGE)   v = agg_b[(long)grow * D_EDGE + (col - D_EDGE)];
        else                         v = vdata[(long)grow * D_NODE + (col - 2 * D_EDGE)];
      }
      xs[row][col] = v;
    }
  }
  __syncthreads();

  const int wave = tid >> 5;
  const int lane = tid & 31;
  const int n0   = wave * 16;        // output-column tile base
  const int mr   = lane & 15;        // A row (lanes 0-15 / 16-31 both map M=0..15)
  const int half = lane >> 4;        // 0: K,K+1 ; 1: K+2,K+3  (per ISA 16x4 layout)
  const int koff = half * 2;

  v8f c = {};
  for (int k = 0; k < D_IN; k += 4) {
    // A fragment: row mr, columns (k+koff, k+koff+1) -> one ds_load_b64 per lane.
    v2f a;
    a.x = xs[mr][k + koff];
    a.y = xs[mr][k + koff + 1];
    // B fragment: rows (k+koff, k+koff+1), column n0+mr (B rows striped across lanes).
    const float* wp = W + (long)(k + koff) * D_OUT + n0 + mr;
    v2f b;
    b.x = wp[0];
    b.y = wp[D_OUT];
    // D = A(16x4) * B(4x16) + C  -- f32 WMMA, 8-arg pattern.
    c = __builtin_amdgcn_wmma_f32_16x16x4_f32(
        /*neg_a=*/false, a, /*neg_b=*/false, b,
        /*c_mod=*/(short)0, c, /*reuse_a=*/false, /*reuse_b=*/false);
  }

  // C/D layout: VGPR r -> row (r + 8*half), col = lane&15 within the tile.
  const int ncol  = n0 + (lane & 15);
  const float bv  = bias[ncol];
  const int mbase = m0 + half * 8;
  if (full_tile) {
#pragma unroll
    for (int r = 0; r < 8; ++r) {
      // out is written once and never re-read: NT store, skip L2 rinse.
      __builtin_nontemporal_store(c[r] + bv, &out[(long)(mbase + r) * D_OUT + ncol]);
    }
  } else {
#pragma unroll
    for (int r = 0; r < 8; ++r) {
      int grow = mbase + r;
      if (grow < n_nodes)
        out[(long)grow * D_OUT + ncol] = c[r] + bv;
    }
  }
}

// ---------------------------------------------------------------------------
// Host-side launcher.
// Inputs (setup_inputs order): vdata, edata_a, edata_b, conn_a, conn_b, W, b
// ---------------------------------------------------------------------------
extern "C" void kernel_launch(void* const* d_in, const int* in_sizes, int n_in,
                              void* d_out, int out_size, void* d_ws, size_t ws_size,
                              hipStream_t stream) {
  const float*     vdata   = (const float*)d_in[0];
  const float*     edata_a = (const float*)d_in[1];
  const float*     edata_b = (const float*)d_in[2];
  const long long* conn_a  = (const long long*)d_in[3];
  const long long* conn_b  = (const long long*)d_in[4];
  const float*     W       = (const float*)d_in[5];
  const float*     bias    = (const float*)d_in[6];
  float*           out     = (float*)d_out;

  const long n_nodes = (long)in_sizes[0] / D_NODE;
  const long n_edges = (long)in_sizes[1] / D_EDGE;

  float* agg_a = (float*)d_ws;
  float* agg_b = agg_a + n_nodes * D_EDGE;

  // 1) zero aggregation buffers (2 * N * 64 f32; divisible by 4)
  const long zn4 = (2 * n_nodes * (long)D_EDGE) / 4;
  zero_f32_kernel<<<2048, 256, 0, stream>>>((float4*)agg_a, zn4);

  // 2) scatter-sum both edge types (receivers = row 1 of conn = conn + n_edges)
  const long total = n_edges * (long)D_EDGE;
  const int  sblocks = (int)((total + 255) / 256);
  scatter_add_kernel<<<sblocks, 256, 0, stream>>>(edata_a, conn_a + n_edges, agg_a, n_edges);
  scatter_add_kernel<<<sblocks, 256, 0, stream>>>(edata_b, conn_b + n_edges, agg_b, n_edges);

  // 3) fused concat + GEMM with f32 WMMA
  const int gblocks = (int)((n_nodes + 15) / 16);
  node_gemm_wmma<<<gblocks, 256, 0, stream>>>(agg_a, agg_b, vdata, W, bias, out,
                                              (int)n_nodes);
}